// TransitionUp_68178310857446
// MI455X (gfx1250) — compile-verified
//
#include <hip/hip_runtime.h>

typedef float v2f __attribute__((ext_vector_type(2)));
typedef float v8f __attribute__((ext_vector_type(8)));

#define CHANNELS 256
#define BN_EPS 1e-5f
#define PCHUNK 4096
#define FLT_BIG 3.402823466e+38f

// ---------------------------------------------------------------------------
// out[n,c] = relu( (X[n,:] . W[c,:] + bias[c] - mean[c]) * g[c]*rsqrt(v[c]+eps)
//                  + beta[c] )  (+ add_src[n,c] if add_src != nullptr)
// One wave computes a 16x64 output tile via V_WMMA_F32_16X16X4_F32.
// f32 WMMA VGPR layout (ISA 7.12.2):
//   A 16x4 : lane m=l&15 holds K = 2*(l>>4)+{0,1}  -> contiguous float2
//   B 4x16 : lane n=l&15 holds K = 2*(l>>4)+{0,1}  (mirror of A)
//   C/D    : VGPR r, lane l -> M = r + 8*(l>>4), N = l&15
// ---------------------------------------------------------------------------
__global__ __launch_bounds__(256) void gemm_bn_relu_kernel(
    const float* __restrict__ X, const float* __restrict__ W,
    const float* __restrict__ bias, const float* __restrict__ gain,
    const float* __restrict__ beta, const float* __restrict__ mean,
    const float* __restrict__ var,
    float* __restrict__ out, const float* __restrict__ add_src, int K)
{
    const int lane  = threadIdx.x & 31;
    const int wave  = threadIdx.x >> 5;
    const int wg    = blockIdx.x * 8 + wave;      // global wave id
    const int row0  = (wg >> 2) * 16;             // 16-row tile
    const int col0  = (wg & 3) * 64;              // 64-col group (4 x 16)
    const int half  = lane >> 4;
    const int idx15 = lane & 15;

    const float* aptr = X + (size_t)(row0 + idx15) * K + 2 * half;
    const float* bptr = W + (size_t)(col0 + idx15) * K + 2 * half;
    const size_t bstride = (size_t)16 * K;        // next 16-col tile of W

    v8f acc[4] = {};
    for (int kk = 0; kk < K; kk += 4) {
        if ((kk & 63) == 0)
            __builtin_prefetch(aptr + kk + 64, 0, 1);   // global_prefetch_b8
        v2f a = *(const v2f*)(aptr + kk);
#pragma unroll
        for (int t = 0; t < 4; ++t) {
            v2f b = *(const v2f*)(bptr + bstride * t + kk);
            acc[t] = __builtin_amdgcn_wmma_f32_16x16x4_f32(
                false, a, false, b, (short)0, acc[t], false, false);
        }
    }

#pragma unroll
    for (int t = 0; t < 4; ++t) {
        const int c = col0 + t * 16 + idx15;
        const float alpha = gain[c] * rsqrtf(var[c] + BN_EPS);
        const float bb    = (bias[c] - mean[c]) * alpha + beta[c];
#pragma unroll
        for (int r = 0; r < 8; ++r) {
            const int rowm = row0 + r + 8 * half;
            float y = acc[t][r] * alpha + bb;
            y = y > 0.0f ? y : 0.0f;
            const size_t o = (size_t)rowm * CHANNELS + c;
            if (add_src) y += add_src[o];
            out[o] = y;
        }
    }
}

// ---------------------------------------------------------------------------
// Per-segment 3-NN + inverse-distance interpolation.
// points_2 staged into LDS as padded float4 slots via the CDNA5 async DMA
// path (GLOBAL_LOAD_ASYNC_TO_LDS_B32, tracked with ASYNCcnt), so the scan
// loop needs a single ds_load_b128 broadcast per candidate.
// ---------------------------------------------------------------------------
__global__ __launch_bounds__(256) void knn_interp_kernel(
    const float* __restrict__ points_1, const float* __restrict__ points_2,
    const int* __restrict__ rs1, const int* __restrict__ rs2,
    const float* __restrict__ f2, float* __restrict__ out, int nseg)
{
    __shared__ float4 sp[PCHUNK];        // padded xyz_ slots, 16B aligned
    __shared__ float  s_w[256 * 3];
    __shared__ int    s_i[256 * 3];

    const int tid    = threadIdx.x;
    const int qfirst = blockIdx.x * 256;
    const int q      = qfirst + tid;

    int seg = 0;
    for (int i = 0; i < nseg; ++i)
        if (qfirst >= rs1[i]) seg = i;
    const int p0 = rs2[seg], p1 = rs2[seg + 1];

    const float qx = points_1[3 * q + 0];
    const float qy = points_1[3 * q + 1];
    const float qz = points_1[3 * q + 2];

    // low 32 bits of a generic pointer to __shared__ = LDS byte offset
    const unsigned lds_base = (unsigned)(size_t)(void*)&sp[0];

    float d0 = FLT_BIG, d1 = FLT_BIG, d2 = FLT_BIG;
    int   i0 = p0, i1 = p0, i2 = p0;

    for (int base = p0; base < p1; base += PCHUNK) {
        const int cnt = min(PCHUNK, p1 - base);
        __syncthreads();
        for (int t = tid; t < cnt; t += 256) {
            const float* src = points_2 + 3 * (size_t)(base + t);
            const unsigned dst = lds_base + 16u * (unsigned)t;
            asm volatile("global_load_async_to_lds_b32 %0, %1, off"
                         :: "v"(dst),      "v"(src)     : "memory");
            asm volatile("global_load_async_to_lds_b32 %0, %1, off offset:4"
                         :: "v"(dst + 4u), "v"(src)     : "memory");
            asm volatile("global_load_async_to_lds_b32 %0, %1, off offset:8"
                         :: "v"(dst + 8u), "v"(src)     : "memory");
        }
        asm volatile("s_wait_asynccnt 0x0" ::: "memory");
        __syncthreads();
#pragma unroll 4
        for (int j = 0; j < cnt; ++j) {
            const float4 p = sp[j];            // one ds_load_b128 broadcast
            const float dx = qx - p.x;
            const float dy = qy - p.y;
            const float dz = qz - p.z;
            const float d  = dx * dx + dy * dy + dz * dz;
            if (d < d2) {                      // strict <  => earliest index wins ties
                if (d < d1) {
                    if (d < d0) { d2 = d1; i2 = i1; d1 = d0; i1 = i0; d0 = d; i0 = base + j; }
                    else        { d2 = d1; i2 = i1; d1 = d;  i1 = base + j; }
                } else          { d2 = d;  i2 = base + j; }
            }
        }
    }

    float w0 = 1.0f / (d0 + 1e-8f);
    float w1 = 1.0f / (d1 + 1e-8f);
    float w2 = 1.0f / (d2 + 1e-8f);
    const float wsum = w0 + w1 + w2;
    w0 /= wsum; w1 /= wsum; w2 /= wsum;

    s_w[tid * 3 + 0] = w0; s_w[tid * 3 + 1] = w1; s_w[tid * 3 + 2] = w2;
    s_i[tid * 3 + 0] = i0; s_i[tid * 3 + 1] = i1; s_i[tid * 3 + 2] = i2;
    __syncthreads();

    // Cooperative write: 256 threads = 256 channels, one query row at a time.
    for (int qq = 0; qq < 256; ++qq) {
        const int   a  = s_i[qq * 3 + 0], b = s_i[qq * 3 + 1], c = s_i[qq * 3 + 2];
        const float wa = s_w[qq * 3 + 0], wb = s_w[qq * 3 + 1], wc = s_w[qq * 3 + 2];
        out[(size_t)(qfirst + qq) * CHANNELS + tid] =
            wa * f2[(size_t)a * CHANNELS + tid] +
            wb * f2[(size_t)b * CHANNELS + tid] +
            wc * f2[(size_t)c * CHANNELS + tid];
    }
}

extern "C" void kernel_launch(void* const* d_in, const int* in_sizes, int n_in,
                              void* d_out, int out_size, void* d_ws, size_t ws_size,
                              hipStream_t stream)
{
    const float* points_1 = (const float*)d_in[0];
    const float* feat_1   = (const float*)d_in[1];
    const int*   rs1      = (const int*)d_in[2];
    const float* points_2 = (const float*)d_in[3];
    const float* feat_2   = (const float*)d_in[4];
    const int*   rs2      = (const int*)d_in[5];
    const float* w1       = (const float*)d_in[6];
    const float* b1       = (const float*)d_in[7];
    const float* g1       = (const float*)d_in[8];
    const float* beta1    = (const float*)d_in[9];
    const float* m1       = (const float*)d_in[10];
    const float* v1       = (const float*)d_in[11];
    const float* w2       = (const float*)d_in[12];
    const float* b2       = (const float*)d_in[13];
    const float* g2       = (const float*)d_in[14];
    const float* beta2    = (const float*)d_in[15];
    const float* m2       = (const float*)d_in[16];
    const float* v2       = (const float*)d_in[17];

    const int N1   = in_sizes[0] / 3;           // 65536
    const int N2   = in_sizes[3] / 3;           // 16384
    const int K2   = in_sizes[4] / N2;          // 512
    const int K1   = in_sizes[1] / N1;          // 256
    const int nseg = in_sizes[2] - 1;           // 4

    float* f2  = (float*)d_ws;                  // 16384*256*4 = 16 MB scratch
    float* out = (float*)d_out;

    // 1) f2 = relu(bn2(feat_2 @ w2^T))
    const int blocks1 = (N2 / 16) * 4 / 8;      // 512
    gemm_bn_relu_kernel<<<blocks1, 256, 0, stream>>>(
        feat_2, w2, b2, g2, beta2, m2, v2, f2, nullptr, K2);

    // 2) interp -> d_out
    knn_interp_kernel<<<N1 / 256, 256, 0, stream>>>(
        points_1, points_2, rs1, rs2, f2, out, nseg);

    // 3) d_out = relu(bn1(feat_1 @ w1^T)) + d_out
    const int blocks3 = (N1 / 16) * 4 / 8;      // 2048
    gemm_bn_relu_kernel<<<blocks3, 256, 0, stream>>>(
        feat_1, w1, b1, g1, beta1, m1, v1, out, out, K1);
}